// GraphAttentionLayer_4629974745450
// MI455X (gfx1250) — compile-verified
//
#include <hip/hip_runtime.h>

#define GAT_B 8
#define GAT_N 1024
#define GAT_D 256
#define GAT_ALPHA 0.2f
#define GAT_NEG_INF -9.0e15f

typedef float v2f __attribute__((ext_vector_type(2)));
typedef float v4f __attribute__((ext_vector_type(4)));
typedef float v8f __attribute__((ext_vector_type(8)));
typedef int gat_v4i __attribute__((vector_size(4 * sizeof(int))));

#define GAT_AS1 __attribute__((address_space(1)))
#define GAT_AS3 __attribute__((address_space(3)))

#if __has_builtin(__builtin_amdgcn_global_load_async_to_lds_b128)
#define GAT_HAVE_ASYNC 1
#else
#define GAT_HAVE_ASYNC 0
#endif

__device__ __forceinline__ v8f wmma_f32(v2f a, v2f b, v8f c) {
  // D = A(16x4,f32) * B(4x16,f32) + C(16x16,f32)
  return __builtin_amdgcn_wmma_f32_16x16x4_f32(
      /*neg_a=*/false, a, /*neg_b=*/false, b,
      /*c_mod=*/(short)0, c, /*reuse_a=*/false, /*reuse_b=*/false);
}

__device__ __forceinline__ void gat_wait_async0() {
#if __has_builtin(__builtin_amdgcn_s_wait_asynccnt)
  __builtin_amdgcn_s_wait_asynccnt(0);
#else
  asm volatile("s_wait_asynccnt 0x0" ::: "memory");
#endif
}

// ---------------------------------------------------------------------------
// Kernel 1: Wh[b,n,f] = sum_d h[b,n,d] * W[f,d]   (flattened M = B*N = 8192)
// One wave computes one 16x16 tile, K-loop of 64 x V_WMMA_F32_16X16X4_F32.
// ---------------------------------------------------------------------------
__global__ __launch_bounds__(256) void gat_gemm_wh(const float* __restrict__ h,
                                                   const float* __restrict__ W,
                                                   float* __restrict__ Wh) {
  const int lane = threadIdx.x & 31;
  const int wave = threadIdx.x >> 5;
  const int t  = blockIdx.x * 8 + wave;      // 0..8191 tiles
  const int m0 = (t >> 4) << 4;              // row tile in flattened B*N
  const int f0 = (t & 15) << 4;              // output-feature tile
  const int fc = lane & 15;                  // M index (A) / N index (B)
  const int kq = (lane >> 4) << 1;           // K sub-offset: 0 or 2

  // A[m][k] = h[m0+m][k],  B[k][f] = W[f0+f][k]  (both rows contiguous in k)
  const float* arow = h + (size_t)(m0 + fc) * GAT_D + kq;
  const float* brow = W + (size_t)(f0 + fc) * GAT_D + kq;

  v8f acc = {};
#pragma unroll 4
  for (int k0 = 0; k0 < GAT_D; k0 += 4) {
    v2f a = *(const v2f*)(arow + k0);
    v2f b = *(const v2f*)(brow + k0);
    acc = wmma_f32(a, b, acc);
  }

  // C layout: VGPR v -> M = v + 8*(lane>=16), N = f0 + (lane&15)
  float* crow = Wh + (size_t)(m0 + ((lane >> 4) << 3)) * GAT_D + f0 + fc;
#pragma unroll
  for (int v = 0; v < 8; ++v) crow[(size_t)v * GAT_D] = acc[v];
}

// ---------------------------------------------------------------------------
// Kernel 2: src[row] = Wh[row,:].a1 ; dst[row] = Wh[row,:].a2  (wave per row)
// ---------------------------------------------------------------------------
__global__ __launch_bounds__(256) void gat_src_dst(const float* __restrict__ Wh,
                                                   const float* __restrict__ a,
                                                   float* __restrict__ srcv,
                                                   float* __restrict__ dstv) {
  const int lane = threadIdx.x & 31;
  const int row  = blockIdx.x * 8 + (threadIdx.x >> 5);
  const float* wr = Wh + (size_t)row * GAT_D;
  float s1 = 0.f, s2 = 0.f;
#pragma unroll
  for (int f = lane; f < GAT_D; f += 32) {
    float w = wr[f];
    s1 = fmaf(w, a[f], s1);
    s2 = fmaf(w, a[GAT_D + f], s2);
  }
  for (int off = 16; off; off >>= 1) {
    s1 += __shfl_xor(s1, off, 32);
    s2 += __shfl_xor(s2, off, 32);
  }
  if (lane == 0) { srcv[row] = s1; dstv[row] = s2; }
}

// ---------------------------------------------------------------------------
// Kernel 3: fused masked softmax + (P @ Wh), one workgroup per (b, 16 rows).
// Pass 1: per-row online softmax stats (no Wh traffic).
// Pass 2: per j-chunk of 32 rows: async-stage Wh(32x256) into LDS while
//         computing P(16x32) with v_exp_f32, then WMMA entirely out of LDS.
// ---------------------------------------------------------------------------
#define GAT_JT 32
#define GAT_WS 272  // WhS stride: 272%64==16 -> K-row pairs hit disjoint banks
#define GAT_PS 36   // P stride: 36*fc mod 64 distinct; 36*d==+-2 mod 64 unsolvable

__global__ __launch_bounds__(256) void gat_attn(const float* __restrict__ Wh,
                                                const int* __restrict__ adj,
                                                const float* __restrict__ srcv,
                                                const float* __restrict__ dstv,
                                                float* __restrict__ out) {
  __shared__ float WhS[GAT_JT * GAT_WS];   // 34.0 KB
  __shared__ float Pb[16 * GAT_PS];        //  2.3 KB
  __shared__ float m_s[16], linv_s[16], src_s[16];

  const int tid  = threadIdx.x;
  const int lane = tid & 31;
  const int wave = tid >> 5;
  const int b  = blockIdx.x >> 6;
  const int i0 = (blockIdx.x & 63) << 4;

  const float* dv   = dstv + b * GAT_N;
  const float* WhB  = Wh + (size_t)b * GAT_N * GAT_D;
  const int*   adjB = adj + ((size_t)b * GAT_N + i0) * GAT_N;

  // ---- pass 1: softmax stats, 2 rows per wave, online (m,l) + lane merge
  for (int rr = 0; rr < 2; ++rr) {
    const int r = wave * 2 + rr;
    const int* arow = adjB + (size_t)r * GAT_N;
    const float si = srcv[b * GAT_N + i0 + r];
    float m = -INFINITY, l = 0.f;
    for (int j = lane; j < GAT_N; j += 32) {
      float x = si + dv[j];
      x = x > 0.f ? x : GAT_ALPHA * x;
      x = arow[j] > 0 ? x : GAT_NEG_INF;
      float mn = fmaxf(m, x);
      l = l * __expf(m - mn) + __expf(x - mn);
      m = mn;
    }
    for (int off = 16; off; off >>= 1) {
      float mo = __shfl_xor(m, off, 32);
      float lo = __shfl_xor(l, off, 32);
      float mn = fmaxf(m, mo);
      l = l * __expf(m - mn) + lo * __expf(mo - mn);
      m = mn;
    }
    if (lane == 0) { m_s[r] = m; linv_s[r] = 1.0f / l; src_s[r] = si; }
  }
  __syncthreads();

  // ---- pass 2: O(16x256) += P(16x32) @ Wh(32x256); wave owns 32 columns
  const int fc = lane & 15;
  const int kq = (lane >> 4) << 1;
  const int f0 = wave << 5;
  const int f1 = f0 + 16;
  v8f acc0 = {}, acc1 = {};

  for (int jc = 0; jc < GAT_N; jc += GAT_JT) {
    // (a) kick off Wh chunk staging: 32 rows x 256 cols = 2048 x 16B
    const float* gsrc = WhB + (size_t)jc * GAT_D;
#if GAT_HAVE_ASYNC
    for (int u = tid; u < (GAT_JT * GAT_D) / 4; u += 256) {
      const int row = u >> 6;            // 64 x 16B units per row
      const int c4  = (u & 63) << 2;     // float column
      __builtin_amdgcn_global_load_async_to_lds_b128(
          (GAT_AS1 gat_v4i*)(gsrc + (size_t)row * GAT_D + c4),
          (GAT_AS3 gat_v4i*)(&WhS[row * GAT_WS + c4]),
          /*offset=*/0, /*cpol=*/0);
    }
#else
    for (int u = tid; u < (GAT_JT * GAT_D) / 4; u += 256) {
      const int row = u >> 6;
      const int c4  = (u & 63) << 2;
      *(v4f*)(&WhS[row * GAT_WS + c4]) = *(const v4f*)(gsrc + (size_t)row * GAT_D + c4);
    }
#endif

    // (b) overlap: compute P tile (16 x 32) with v_exp_f32
    for (int e = tid; e < 16 * GAT_JT; e += 256) {
      const int r  = e >> 5;
      const int jl = e & (GAT_JT - 1);
      const int j  = jc + jl;
      float x = src_s[r] + dv[j];
      x = x > 0.f ? x : GAT_ALPHA * x;
      x = adjB[(size_t)r * GAT_N + j] > 0 ? x : GAT_NEG_INF;
      Pb[r * GAT_PS + jl] = __expf(x - m_s[r]) * linv_s[r];
    }

#if GAT_HAVE_ASYNC
    gat_wait_async0();
#endif
    __syncthreads();

    // (c) 8 K-steps of fp32 WMMA, A and B frags both from LDS
#pragma unroll
    for (int k0 = 0; k0 < GAT_JT; k0 += 4) {
      v2f a = *(const v2f*)(&Pb[fc * GAT_PS + k0 + kq]);
      const float* w0 = &WhS[(k0 + kq) * GAT_WS + fc];
      v2f b0, b1;
      b0.x = w0[f0];          b0.y = w0[GAT_WS + f0];
      b1.x = w0[f1];          b1.y = w0[GAT_WS + f1];
      acc0 = wmma_f32(a, b0, acc0);
      acc1 = wmma_f32(a, b1, acc1);
    }
    __syncthreads();
  }

  float* orow = out + ((size_t)b * GAT_N + i0 + ((lane >> 4) << 3)) * GAT_D + fc;
#pragma unroll
  for (int v = 0; v < 8; ++v) {
    orow[(size_t)v * GAT_D + f0] = acc0[v];
    orow[(size_t)v * GAT_D + f1] = acc1[v];
  }
}

// ---------------------------------------------------------------------------
extern "C" void kernel_launch(void* const* d_in, const int* in_sizes, int n_in,
                              void* d_out, int out_size, void* d_ws, size_t ws_size,
                              hipStream_t stream) {
  (void)in_sizes; (void)n_in; (void)out_size; (void)ws_size;
  const float* h   = (const float*)d_in[0];
  const int*   adj = (const int*)d_in[1];
  const float* W   = (const float*)d_in[2];
  const float* a   = (const float*)d_in[3];
  float* out = (float*)d_out;

  float* Wh   = (float*)d_ws;                          // 8 MB
  float* srcv = Wh + (size_t)GAT_B * GAT_N * GAT_D;    // 32 KB
  float* dstv = srcv + GAT_B * GAT_N;                  // 32 KB

  gat_gemm_wh<<<1024, 256, 0, stream>>>(h, W, Wh);
  gat_src_dst<<<1024, 256, 0, stream>>>(Wh, a, srcv, dstv);
  gat_attn<<<GAT_B * (GAT_N / 16), 256, 0, stream>>>(Wh, adj, srcv, dstv, out);
}